// SparseLayer_56281251447203
// MI455X (gfx1250) — compile-verified
//
#include <hip/hip_runtime.h>
#include <math.h>

// MI455X / gfx1250 implementation.
// One workgroup (384 threads = 12 wave32) per (b,k) group; 1024 groups total.
// Column-sum normalization is done on the matrix unit via chained
// V_WMMA_F32_16X16X4_F32 (A = all-ones -> D rows = column sums of B),
// which is layout-robust: it only relies on the documented lane->N mapping
// of B/C/D, not on the K-slot ordering inside a column.

typedef float v2f __attribute__((ext_vector_type(2)));
typedef float v8f __attribute__((ext_vector_type(8)));

#define N_TOT   4096
#define C_      32
#define K_      128
#define PPC     12          // points per component: 4 corners + 4 sampled + 4 rr
#define NPTS    (C_ * PPC)  // 384
#define IN_SZ   1024
#define OUT_SZ  1024
#define EPSF    1e-6f

__global__ __launch_bounds__(NPTS) void sparse_layer_kernel(
    const float* __restrict__ x,       // (b, 1024)
    const float* __restrict__ means,   // (b, 4096, 2)
    const float* __restrict__ sigmas,  // (b, 4096, 2)
    const float* __restrict__ values,  // (b, 4096)
    const float* __restrict__ su,      // (b, 128, 32, 4, 2)
    const float* __restrict__ rru,     // (b, 128, 32, 4, 2)
    float* __restrict__ out)           // (b, 1024)
{
    __shared__ float s_mean[C_][2];
    __shared__ float s_invsq[C_][2];
    __shared__ float s_val[C_];
    __shared__ float s_w[C_];
    __shared__ float s_colsum[C_];
    __shared__ int   s_pt0[NPTS];
    __shared__ int   s_pt1[NPTS];
    __shared__ int   s_key[NPTS];
    __shared__ int   s_dup[NPTS];
    __shared__ float s_props[C_][NPTS];   // comp-major: 48 KB

    const int tid = threadIdx.x;          // 0..383 == point index p
    const int bk  = blockIdx.x;           // 0..1023
    const int b   = bk >> 7;              // / 128
    const int k   = bk & (K_ - 1);

    // ---- Stage 0: per-component parameters into LDS ----
    if (tid < C_) {
        const int  n    = k * C_ + tid;
        const long base = ((long)b * N_TOT + n) * 2;
        s_mean[tid][0]  = means[base + 0];
        s_mean[tid][1]  = means[base + 1];
        s_invsq[tid][0] = 1.0f / (EPSF + sigmas[base + 0]);
        s_invsq[tid][1] = 1.0f / (EPSF + sigmas[base + 1]);
        s_val[tid]      = values[(long)b * N_TOT + n];
    }
    __syncthreads();

    // ---- Stage 1: generate the 384 integer points ----
    {
        const int p = tid;
        const int c = p / PPC;
        const int j = p - c * PPC;            // 0..11
        const float m0 = s_mean[c][0];
        const float m1 = s_mean[c][1];
        int p0, p1;
        if (j < 4) {
            // itertools.product order: j=0 (floor,floor) 1 (floor,ceil)
            //                          2 (ceil,floor)    3 (ceil,ceil)
            p0 = (j <= 1)       ? (int)floorf(m0) : (int)ceilf(m0);
            p1 = ((j & 1) == 0) ? (int)floorf(m1) : (int)ceilf(m1);
        } else if (j < 8) {
            const long base = ((((long)b * K_ + k) * C_ + c) * 4 + (j - 4)) * 2;
            p0 = (int)floorf(su[base + 0] * (1.0f - EPSF) * (float)OUT_SZ);
            p1 = (int)floorf(su[base + 1] * (1.0f - EPSF) * (float)IN_SZ);
        } else {
            const long base = ((((long)b * K_ + k) * C_ + c) * 4 + (j - 8)) * 2;
            const float mn0 = rintf(m0);      // round-half-even == jnp.round
            const float mn1 = rintf(m1);
            float lo0 = mn0 - 8.0f;
            if (lo0 < 0.0f)           lo0 = 0.0f;
            if (mn0 + 8.0f > 1024.0f) lo0 = 1008.0f;   // rng - rrng
            float lo1 = mn1 - 8.0f;
            if (lo1 < 0.0f)           lo1 = 0.0f;
            if (mn1 + 8.0f > 1024.0f) lo1 = 1008.0f;
            p0 = (int)floorf(rru[base + 0] * (1.0f - EPSF) * 16.0f + lo0);
            p1 = (int)floorf(rru[base + 1] * (1.0f - EPSF) * 16.0f + lo1);
        }
        s_pt0[p] = p0;
        s_pt1[p] = p1;
        s_key[p] = p0 * IN_SZ + p1;
    }
    __syncthreads();

    // ---- Stage 2: duplicate marking ----
    // Stable-argsort semantics: dup iff an EARLIER point has the same key.
    {
        const int p   = tid;
        const int key = s_key[p];
        int dup = 0;
        for (int q = 0; q < p; ++q)           // broadcast LDS reads
            dup |= (s_key[q] == key);
        s_dup[p] = dup;
    }
    __syncthreads();

    // ---- Stage 3: Gaussian densities (exact form, no cancellation) ----
    {
        const int   p   = tid;
        const float px0 = (float)s_pt0[p];
        const float px1 = (float)s_pt1[p];
        const int   dup = s_dup[p];
        #pragma unroll 4
        for (int c = 0; c < C_; ++c) {
            const float d0 = px0 - s_mean[c][0];
            const float d1 = px1 - s_mean[c][1];
            const float s  = d0 * d0 * s_invsq[c][0] + d1 * d1 * s_invsq[c][1];
            s_props[c][p]  = dup ? 0.0f : expf(-0.5f * s);
        }
    }
    __syncthreads();

    // ---- Stage 4: column sums via WMMA (waves 0,1 handle 16 comps each) ----
    // D = ones(16x4) x B(4x16) + C  ==> every row of D is colsum(B).
    // Per call each lane feeds 2 f32 of its column c = lane&15; lanes L and
    // L+16 jointly supply the 4 K-slots of column c (sum is K-order invariant).
    {
        const int wave = tid >> 5;
        const int lane = tid & 31;
        if (wave < 2) {
            const int c  = wave * 16 + (lane & 15);
            const int hi = lane >> 4;
            v2f a;  a.x = 1.0f;  a.y = 1.0f;      // all-ones A: layout-proof
            v8f acc = {};
            #pragma unroll 4
            for (int t = 0; t < NPTS / 4; ++t) {  // 96 chained WMMAs
                const int p = 4 * t + 2 * hi;
                v2f bv;
                bv.x = s_props[c][p];
                bv.y = s_props[c][p + 1];
                acc = __builtin_amdgcn_wmma_f32_16x16x4_f32(
                        /*neg_a=*/false, a, /*neg_b=*/false, bv,
                        /*c_mod=*/(short)0, acc,
                        /*reuse_a=*/false, /*reuse_b=*/false);
            }
            if (lane < 16) s_colsum[c] = acc[0];  // D: VGPR0,lane n -> M=0,N=n
        }
    }
    __syncthreads();

    if (tid < C_) s_w[tid] = s_val[tid] / s_colsum[tid];
    __syncthreads();

    // ---- Stage 5: vals = props . w ; scatter-add into out ----
    {
        const int p = tid;
        float v = 0.0f;
        #pragma unroll 8
        for (int c = 0; c < C_; ++c)
            v += s_props[c][p] * s_w[c];
        const float contrib = v * x[(long)b * IN_SZ + s_pt1[p]];
        atomicAdd(&out[(long)b * OUT_SZ + s_pt0[p]], contrib);
    }
}

extern "C" void kernel_launch(void* const* d_in, const int* in_sizes, int n_in,
                              void* d_out, int out_size, void* d_ws, size_t ws_size,
                              hipStream_t stream) {
    (void)n_in; (void)d_ws; (void)ws_size;
    const float* x      = (const float*)d_in[0];
    const float* means  = (const float*)d_in[1];
    const float* sigmas = (const float*)d_in[2];
    const float* values = (const float*)d_in[3];
    const float* su     = (const float*)d_in[4];
    const float* rru    = (const float*)d_in[5];
    float* out = (float*)d_out;

    const int b = in_sizes[3] / N_TOT;        // values is (b, 4096) -> b = 8

    // Zero the scatter target (graph-capturable async memset).
    hipMemsetAsync(d_out, 0, (size_t)out_size * sizeof(float), stream);

    dim3 grid(b * K_);                         // 1024 groups
    dim3 block(NPTS);                          // 384 threads = 12 wave32
    sparse_layer_kernel<<<grid, block, 0, stream>>>(x, means, sigmas, values,
                                                    su, rru, out);
}